// GCNClassifierLinear_66340064854352
// MI455X (gfx1250) — compile-verified
//
#include <hip/hip_runtime.h>

#define NN 100000   // nodes
#define NE 1600000  // edges
#define HD 128      // hidden channels
#define NG 64       // graphs
#define OC 64       // fc1 out channels

typedef float v2f __attribute__((ext_vector_type(2)));
typedef float v8f __attribute__((ext_vector_type(8)));

__device__ __forceinline__ unsigned f2ord(float f) {
  unsigned u = __float_as_uint(f);
  return (u & 0x80000000u) ? ~u : (u | 0x80000000u);
}
__device__ __forceinline__ float ord2f(unsigned u) {
  return (u & 0x80000000u) ? __uint_as_float(u & 0x7fffffffu) : __uint_as_float(~u);
}

// ---- init: buf[i] = bias[i % HD]  (fuses the "+ b" of segment_sum into scatter target)
__global__ __launch_bounds__(256) void fill_bias_k(float* __restrict__ buf,
                                                   const float* __restrict__ bias, int n) {
  int t = blockIdx.x * 256 + threadIdx.x;
  if (t < n) buf[t] = bias[t & (HD - 1)];
}

__global__ __launch_bounds__(256) void pool_init_k(float* __restrict__ sums,
                                                   unsigned* __restrict__ maxs,
                                                   float* __restrict__ counts) {
  int t = blockIdx.x * 256 + threadIdx.x;
  if (t < NG * HD) { sums[t] = 0.f; maxs[t] = 0x007FFFFFu; }  // f2ord(-inf)
  if (t < NG) counts[t] = 0.f;
}

// ---- out[r,:] = A[idx ? idx[r] : r, :] @ W   via V_WMMA_F32_16X16X4_F32
// 256 threads = 8 waves; each wave owns 16 output rows x all 128 cols (8 col tiles).
// W staged in LDS K-pair-interleaved: sW[(k/2)*256 + col*2 + (k&1)] = W[k][col]
// so each B fragment (W[k][col], W[k+1][col]) is one contiguous ds_load_b64.
__global__ __launch_bounds__(256) void gemm_wmma_k(const float* __restrict__ A,
                                                   const int* __restrict__ idx,
                                                   const float* __restrict__ W,
                                                   float* __restrict__ out, int nrows) {
  __shared__ float sW[HD * HD];
  for (int i = threadIdx.x; i < HD * HD; i += 256) {
    int k = i >> 7;          // i / HD
    int col = i & (HD - 1);  // i % HD
    sW[(k >> 1) * 256 + col * 2 + (k & 1)] = W[i];
  }
  __syncthreads();

  const int wave = threadIdx.x >> 5;
  const int lane = threadIdx.x & 31;
  const int half = lane >> 4;   // lanes 16-31 hold K+2/K+3
  const int lr   = lane & 15;

  const int rowBase = blockIdx.x * 128 + wave * 16;
  if (rowBase >= nrows) return;  // wave-uniform

  // A fragment source row for this lane (clamped; masked at store instead of EXEC)
  const int row  = rowBase + lr;
  const int srow = (row < nrows) ? (idx ? idx[row] : row) : 0;
  const float* __restrict__ arow = A + (size_t)srow * HD;

  v8f acc[8] = {};

  for (int k0 = 0; k0 < HD; k0 += 4) {
    const int k = k0 + 2 * half;
    v2f a = *(const v2f*)(arow + k);               // global_load_b64
    const float* bbase = &sW[(k >> 1) * 256 + lr * 2];
#pragma unroll
    for (int ct = 0; ct < 8; ++ct) {
      v2f b = *(const v2f*)(bbase + ct * 32);      // ds_load_b64, contiguous pair
      acc[ct] = __builtin_amdgcn_wmma_f32_16x16x4_f32(
          false, a, false, b, (short)0, acc[ct], false, false);
    }
  }

#pragma unroll
  for (int ct = 0; ct < 8; ++ct) {
#pragma unroll
    for (int r = 0; r < 8; ++r) {
      int m = rowBase + r + 8 * half;  // C/D layout: VGPR r -> M=r (lanes 0-15), M=r+8 (16-31)
      if (m < nrows) out[(size_t)m * HD + ct * 16 + lr] = acc[ct][r];
    }
  }
}

// ---- per-edge gather + weight + scatter-add: one lane per (edge, 4 channels)
__global__ __launch_bounds__(256) void edge_scatter_k(const int* __restrict__ ei,
                                                      const float* __restrict__ ew,
                                                      const float* __restrict__ hw,
                                                      float* __restrict__ out) {
  long long t = (long long)blockIdx.x * 256 + threadIdx.x;
  if (t >= (long long)NE * 32) return;
  int e  = (int)(t >> 5);
  int c4 = ((int)t & 31) * 4;
  int src = ei[e];
  int dst = ei[NE + e];
  float w = ew[e];
  const float4 v = *(const float4*)(hw + (size_t)src * HD + c4);
  float* o = out + (size_t)dst * HD + c4;
  unsafeAtomicAdd(o + 0, v.x * w);   // -> GLOBAL_ATOMIC_ADD_F32 (no return)
  unsafeAtomicAdd(o + 1, v.y * w);
  unsafeAtomicAdd(o + 2, v.z * w);
  unsafeAtomicAdd(o + 3, v.w * w);
}

// ---- global mean/max pool: one lane per (node, 4 channels)
__global__ __launch_bounds__(256) void pool_k(const float* __restrict__ h,
                                              const int* __restrict__ batch,
                                              float* __restrict__ sums,
                                              unsigned* __restrict__ maxs,
                                              float* __restrict__ counts) {
  long long t = (long long)blockIdx.x * 256 + threadIdx.x;
  if (t >= (long long)NN * 32) return;
  int node = (int)(t >> 5);
  int c4   = ((int)t & 31) * 4;
  int g = batch[node];
  const float4 v = *(const float4*)(h + (size_t)node * HD + c4);
  float*    s = sums + g * HD + c4;
  unsigned* m = maxs + g * HD + c4;
  unsafeAtomicAdd(s + 0, v.x);  atomicMax(m + 0, f2ord(v.x));
  unsafeAtomicAdd(s + 1, v.y);  atomicMax(m + 1, f2ord(v.y));
  unsafeAtomicAdd(s + 2, v.z);  atomicMax(m + 2, f2ord(v.z));
  unsafeAtomicAdd(s + 3, v.w);  atomicMax(m + 3, f2ord(v.w));
  if (((int)t & 31) == 0) unsafeAtomicAdd(counts + g, 1.f);
}

// ---- gfeat[g] = [ sum/count , decoded max ]
__global__ __launch_bounds__(256) void gfeat_k(const float* __restrict__ sums,
                                               const unsigned* __restrict__ maxs,
                                               const float* __restrict__ counts,
                                               float* __restrict__ gfeat) {
  int t = blockIdx.x * 256 + threadIdx.x;
  if (t >= NG * 2 * HD) return;
  int g = t / (2 * HD);
  int c = t % (2 * HD);
  float r;
  if (c < HD) r = sums[g * HD + c] / fmaxf(counts[g], 1.f);
  else        r = ord2f(maxs[g * HD + (c - HD)]);
  gfeat[t] = r;
}

// ---- tiny FCs (64x256x64 then 64x64x2): VALU is plenty
__global__ __launch_bounds__(256) void fc1_k(const float* __restrict__ gfeat,
                                             const float* __restrict__ w,
                                             const float* __restrict__ b,
                                             float* __restrict__ hid) {
  int t = blockIdx.x * 256 + threadIdx.x;  // NG*OC = 4096
  if (t >= NG * OC) return;
  int g = t >> 6, o = t & 63;
  float s = b[o];
  const float* gv = gfeat + g * 2 * HD;
  const float* wv = w + o * 2 * HD;
  for (int i = 0; i < 2 * HD; ++i) s += gv[i] * wv[i];
  hid[t] = s;
}

__global__ void fc2_k(const float* __restrict__ hid, const float* __restrict__ w,
                      const float* __restrict__ b, float* __restrict__ out) {
  int t = blockIdx.x * blockDim.x + threadIdx.x;  // NG*2 = 128
  if (t >= NG * 2) return;
  int g = t >> 1, j = t & 1;
  float s = b[j];
  const float* hv = hid + g * OC;
  const float* wv = w + j * OC;
  for (int i = 0; i < OC; ++i) s += hv[i] * wv[i];
  out[t] = s;
}

extern "C" void kernel_launch(void* const* d_in, const int* in_sizes, int n_in,
                              void* d_out, int out_size, void* d_ws, size_t ws_size,
                              hipStream_t stream) {
  const int*   x     = (const int*)  d_in[0];
  const int*   ei    = (const int*)  d_in[1];   // [2, E]
  const float* ew    = (const float*)d_in[2];
  const int*   batch = (const int*)  d_in[3];
  const float* emb   = (const float*)d_in[4];
  const float* W0    = (const float*)d_in[5];
  const float* b0    = (const float*)d_in[6];
  const float* W1    = (const float*)d_in[7];
  const float* b1    = (const float*)d_in[8];
  const float* fc1w  = (const float*)d_in[9];
  const float* fc1b  = (const float*)d_in[10];
  const float* fc2w  = (const float*)d_in[11];
  const float* fc2b  = (const float*)d_in[12];

  const size_t NH = (size_t)NN * HD;  // 12.8M floats
  float*    bufA   = (float*)d_ws;              // layer-2 scatter target / final h
  float*    bufB   = bufA + NH;                 // layer-1 scatter target / layer-2 input
  float*    bufC   = bufB + NH;                 // hw (GEMM output)
  float*    sums   = bufC + NH;                 // [G,H]
  unsigned* maxs   = (unsigned*)(sums + NG * HD);
  float*    counts = (float*)(maxs + NG * HD);  // [G]
  float*    gfeat  = counts + NG;               // [G, 2H]
  float*    hid    = gfeat + NG * 2 * HD;       // [G, OC]

  const int nhBlocks   = (int)((NH + 255) / 256);
  const int gemmBlocks = (NN + 127) / 128;
  const int escBlocks  = (int)(((long long)NE * 32 + 255) / 256);
  const int poolBlocks = (int)(((long long)NN * 32 + 255) / 256);

  // Layer 1: h1 = scatter_add(dst, (emb[x] @ W0)[src] * ew) + b0
  fill_bias_k<<<nhBlocks, 256, 0, stream>>>(bufB, b0, (int)NH);
  gemm_wmma_k<<<gemmBlocks, 256, 0, stream>>>(emb, x, W0, bufC, NN);
  edge_scatter_k<<<escBlocks, 256, 0, stream>>>(ei, ew, bufC, bufB);

  // Layer 2
  fill_bias_k<<<nhBlocks, 256, 0, stream>>>(bufA, b1, (int)NH);
  gemm_wmma_k<<<gemmBlocks, 256, 0, stream>>>(bufB, nullptr, W1, bufC, NN);
  edge_scatter_k<<<escBlocks, 256, 0, stream>>>(ei, ew, bufC, bufA);

  // Pooling + heads
  pool_init_k<<<(NG * HD + 255) / 256, 256, 0, stream>>>(sums, maxs, counts);
  pool_k<<<poolBlocks, 256, 0, stream>>>(bufA, batch, sums, maxs, counts);
  gfeat_k<<<(NG * 2 * HD + 255) / 256, 256, 0, stream>>>(sums, maxs, counts, gfeat);
  fc1_k<<<(NG * OC + 255) / 256, 256, 0, stream>>>(gfeat, fc1w, fc1b, hid);
  fc2_k<<<1, NG * 2, 0, stream>>>(hid, fc2w, fc2b, (float*)d_out);
}